// GINEncoder_3882650435628
// MI455X (gfx1250) — compile-verified
//
#include <hip/hip_runtime.h>
#include <hip/hip_bf16.h>

typedef float v2f __attribute__((ext_vector_type(2)));
typedef float v8f __attribute__((ext_vector_type(8)));

#define HID 128
#define INDIM 100

// ---------------------------------------------------------------- zero fill
__global__ void zero_kernel(float* __restrict__ p, long long n) {
    long long i = (long long)blockIdx.x * blockDim.x + threadIdx.x;
    if (i < n) p[i] = 0.0f;
}

// ---------------------------------------------------------------- scatter-add (segment_sum of x[src] into agg[dst])
// One thread per (edge, float4-chunk). D4 = D/4 compile-time so idx/D4 folds to mul-shift.
template <int D4>
__global__ void scatter_add_kernel(const float* __restrict__ x,
                                   const int* __restrict__ src,
                                   const int* __restrict__ dst,
                                   float* __restrict__ agg,
                                   long long nEdges) {
    long long idx = (long long)blockIdx.x * blockDim.x + threadIdx.x;
    long long total = nEdges * D4;
    if (idx >= total) return;
    int e = (int)(idx / D4);
    int c = (int)(idx % D4);
    int s = src[e];
    int d = dst[e];
    const float4 v = ((const float4*)x)[(long long)s * D4 + c];
    float* out = agg + (long long)d * (D4 * 4) + c * 4;
    unsafeAtomicAdd(out + 0, v.x);
    unsafeAtomicAdd(out + 1, v.y);
    unsafeAtomicAdd(out + 2, v.z);
    unsafeAtomicAdd(out + 3, v.w);
}

// ---------------------------------------------------------------- WMMA GEMM: out[M x 128] = f(A)[M x K] * W[K x 128] + bias
// BNRELU=false: f(A) = A + aux0 (row-wise elementwise add, aux0 is [M x K])
// BNRELU=true : f(A) = relu(A * aux0[k] + aux1[k]) (per-column BN scale/shift)
// One wave per 16-row strip; 8 v8f accumulators cover the full N=128 width.
template <int K, bool BNRELU>
__global__ __launch_bounds__(128) void gin_gemm_kernel(
    const float* __restrict__ A, const float* __restrict__ aux0,
    const float* __restrict__ aux1, const float* __restrict__ W,
    const float* __restrict__ bias, float* __restrict__ out, int M) {
    const int lane  = threadIdx.x & 31;
    const int wave  = threadIdx.x >> 5;
    const int m0    = (blockIdx.x * 4 + wave) * 16;
    if (m0 >= M) return;                    // wave-uniform: EXEC stays all-ones for WMMA
    const int lrow  = lane & 15;
    const int khalf = lane >> 4;            // 0: K=0,1 ; 1: K=2,3 (f32 16x4 A layout)
    const int row   = m0 + lrow;

    v8f acc[8] = {};

    const float* arow   = A + (long long)row * K;
    const float* auxrow = BNRELU ? nullptr : (aux0 + (long long)row * K);

    for (int k = 0; k < K; k += 4) {
        const int ka = k + 2 * khalf;
        v2f a;
        if (BNRELU) {
            a.x = fmaxf(0.0f, arow[ka + 0] * aux0[ka + 0] + aux1[ka + 0]);
            a.y = fmaxf(0.0f, arow[ka + 1] * aux0[ka + 1] + aux1[ka + 1]);
        } else {
            a.x = arow[ka + 0] + auxrow[ka + 0];
            a.y = arow[ka + 1] + auxrow[ka + 1];
        }
        const float* w0 = W + (long long)(ka + 0) * HID + lrow;
        const float* w1 = W + (long long)(ka + 1) * HID + lrow;
#pragma unroll
        for (int t = 0; t < 8; ++t) {
            v2f b;
            b.x = w0[t * 16];               // B 4x16 layout: lanes0-15 K=0/1, lanes16-31 K=2/3
            b.y = w1[t * 16];
            acc[t] = __builtin_amdgcn_wmma_f32_16x16x4_f32(
                false, a, false, b, (short)0, acc[t], false, false);
        }
    }

#pragma unroll
    for (int t = 0; t < 8; ++t) {
        const int n  = t * 16 + lrow;
        const float bn = bias[n];
#pragma unroll
        for (int r = 0; r < 8; ++r) {
            out[(long long)(m0 + r + 8 * khalf) * HID + n] = acc[t][r] + bn;
        }
    }
}

// ---------------------------------------------------------------- column sums / sum-of-squares for BatchNorm stats
__global__ __launch_bounds__(HID) void colstats_kernel(const float* __restrict__ h,
                                                       float* __restrict__ sums,
                                                       float* __restrict__ sumsq,
                                                       int M) {
    const int c = threadIdx.x;              // 128 threads = one per column
    float s = 0.0f, s2 = 0.0f;
    for (int r = blockIdx.x; r < M; r += gridDim.x) {
        float v = h[(long long)r * HID + c];
        s += v;
        s2 += v * v;
    }
    unsafeAtomicAdd(&sums[c], s);
    unsafeAtomicAdd(&sumsq[c], s2);
}

__global__ __launch_bounds__(HID) void bnfinalize_kernel(const float* __restrict__ sums,
                                                         const float* __restrict__ sumsq,
                                                         const float* __restrict__ gamma,
                                                         const float* __restrict__ beta,
                                                         float* __restrict__ scale,
                                                         float* __restrict__ shift,
                                                         float invM) {
    const int c = threadIdx.x;
    float mean = sums[c] * invM;
    float var  = sumsq[c] * invM - mean * mean;
    float sc   = gamma[c] * rsqrtf(var + 1e-5f);
    scale[c] = sc;
    shift[c] = beta[c] - mean * sc;
}

// ---------------------------------------------------------------- launch
extern "C" void kernel_launch(void* const* d_in, const int* in_sizes, int n_in,
                              void* d_out, int out_size, void* d_ws, size_t ws_size,
                              hipStream_t stream) {
    const float* x      = (const float*)d_in[0];
    const int*   edges  = (const int*)d_in[1];
    const float* W1a    = (const float*)d_in[2];
    const float* b1a    = (const float*)d_in[3];
    const float* gamma1 = (const float*)d_in[4];
    const float* beta1  = (const float*)d_in[5];
    const float* W1b    = (const float*)d_in[6];
    const float* b1b    = (const float*)d_in[7];
    const float* W2a    = (const float*)d_in[8];
    const float* b2a    = (const float*)d_in[9];
    const float* gamma2 = (const float*)d_in[10];
    const float* beta2  = (const float*)d_in[11];
    const float* W2b    = (const float*)d_in[12];
    const float* b2b    = (const float*)d_in[13];

    const int M = in_sizes[0] / INDIM;      // 50000 nodes
    const long long E = in_sizes[1] / 2;    // 800000 edges
    const int* src = edges;
    const int* dst = edges + E;

    // workspace layout (all float32)
    float* ws    = (float*)d_ws;
    float* agg   = ws;                               // M * 128 (layer1 uses first M*100)
    float* h     = agg + (long long)M * HID;         // M * 128
    float* x2    = h + (long long)M * HID;           // M * 128 (layer-1 output)
    float* sums  = x2 + (long long)M * HID;          // 128
    float* sumsq = sums + HID;                       // 128
    float* scale = sumsq + HID;                      // 128
    float* shift = scale + HID;                      // 128

    const dim3 blk256(256), blk128(128);
    const int gemmBlocks = (M / 16 + 3) / 4;         // 4 waves per block, 16 rows per wave
    const float invM = 1.0f / (float)M;

    // ======================= Layer 1 (K = 100) =======================
    {
        long long nAgg = (long long)M * INDIM;
        zero_kernel<<<(unsigned)((nAgg + 255) / 256), blk256, 0, stream>>>(agg, nAgg);
        zero_kernel<<<1, blk256, 0, stream>>>(sums, 2 * HID);

        long long total = E * (INDIM / 4);
        scatter_add_kernel<INDIM / 4><<<(unsigned)((total + 255) / 256), blk256, 0, stream>>>(
            x, src, dst, agg, E);

        // h = (x + agg) @ W1a + b1a
        gin_gemm_kernel<INDIM, false><<<gemmBlocks, blk128, 0, stream>>>(
            x, agg, nullptr, W1a, b1a, h, M);

        colstats_kernel<<<256, blk128, 0, stream>>>(h, sums, sumsq, M);
        bnfinalize_kernel<<<1, blk128, 0, stream>>>(sums, sumsq, gamma1, beta1, scale, shift, invM);

        // x2 = relu(BN(h)) @ W1b + b1b   (BN+ReLU fused into A-operand load)
        gin_gemm_kernel<HID, true><<<gemmBlocks, blk128, 0, stream>>>(
            h, scale, shift, W1b, b1b, x2, M);
    }

    // ======================= Layer 2 (K = 128) =======================
    {
        long long nAgg = (long long)M * HID;
        zero_kernel<<<(unsigned)((nAgg + 255) / 256), blk256, 0, stream>>>(agg, nAgg);
        zero_kernel<<<1, blk256, 0, stream>>>(sums, 2 * HID);

        long long total = E * (HID / 4);
        scatter_add_kernel<HID / 4><<<(unsigned)((total + 255) / 256), blk256, 0, stream>>>(
            x2, src, dst, agg, E);

        // h = (x2 + agg) @ W2a + b2a
        gin_gemm_kernel<HID, false><<<gemmBlocks, blk128, 0, stream>>>(
            x2, agg, nullptr, W2a, b2a, h, M);

        colstats_kernel<<<256, blk128, 0, stream>>>(h, sums, sumsq, M);
        bnfinalize_kernel<<<1, blk128, 0, stream>>>(sums, sumsq, gamma2, beta2, scale, shift, invM);

        // out = relu(BN(h)) @ W2b + b2b
        gin_gemm_kernel<HID, true><<<gemmBlocks, blk128, 0, stream>>>(
            h, scale, shift, W2b, b2b, (float*)d_out, M);
    }
}